// GCNConvBlock2_10161892622614
// MI455X (gfx1250) — compile-verified
//
#include <hip/hip_runtime.h>

typedef __attribute__((ext_vector_type(2))) float v2f;
typedef __attribute__((ext_vector_type(8))) float v8f;

#define D_FEAT 128
#define XS_STRIDE 132   // 16-row X tile, padded: conflict-free ds_load_b64 for A operand

// ---------------------------------------------------------------------------
// Degree init: every node starts at 1.0 (self loop). Also re-initializes the
// workspace each call (harness does not re-poison/zero between replays).
// ---------------------------------------------------------------------------
__global__ void deg_init_kernel(float* __restrict__ deg, int n) {
    int i = blockIdx.x * blockDim.x + threadIdx.x;
    if (i < n) deg[i] = 1.0f;
}

// One thread per edge: deg[dst[e]] += 1.0 (hardware f32 atomic; counts exact in f32)
__global__ void deg_count_kernel(const int* __restrict__ dst, float* __restrict__ deg, int e) {
    int i = blockIdx.x * blockDim.x + threadIdx.x;
    if (i < e) unsafeAtomicAdd(&deg[dst[i]], 1.0f);
}

// deg -> deg^{-1/2} in place (deg >= 1 always, so no zero guard needed)
__global__ void deg_rsqrt_kernel(float* __restrict__ deg, int n) {
    int i = blockIdx.x * blockDim.x + threadIdx.x;
    if (i < n) deg[i] = rsqrtf(deg[i]);
}

// Wt[n][k] = W[k][n]: makes the WMMA B fragment a contiguous 8-byte load.
__global__ void w_transpose_kernel(const float* __restrict__ W, float* __restrict__ Wt) {
    int t = blockIdx.x * blockDim.x + threadIdx.x;   // 16384 threads
    int k = t >> 7;          // row of W
    int nc = t & 127;        // col of W
    Wt[nc * D_FEAT + k] = W[t];
}

// ---------------------------------------------------------------------------
// H = X @ W via V_WMMA_F32_16X16X4_F32.
// Block = 256 threads = 8 waves; block owns a 16-row M tile; wave w owns
// N-columns [16w, 16w+16). K loop: 32 x wmma 16x16x4.
// X tile staged memory->LDS with gfx1250 async loads (ASYNCcnt).
// ---------------------------------------------------------------------------
__launch_bounds__(256)
__global__ void gcn_gemm_wmma_kernel(const float* __restrict__ X,
                                     const float* __restrict__ Wt,
                                     float* __restrict__ H, int n) {
    __shared__ float Xs[16 * XS_STRIDE];  // 2112 floats = 8448 B

    const int t = threadIdx.x;
    const int mBase = blockIdx.x * 16;

    // Stage X tile (16 x 128 f32, 8 KB) with async global->LDS B128 copies.
    // Low 32 bits of a generic shared-memory pointer == LDS byte offset.
    #pragma unroll
    for (int i = 0; i < 2; ++i) {
        int idx = (t + i * 256) * 4;      // flat float index into 16x128 tile
        int row = idx >> 7;               // /128
        int col = idx & 127;
        if (mBase + row < n) {
            const float* gsrc = X + (size_t)(mBase + row) * D_FEAT + col;
            unsigned lds_off = (unsigned)(size_t)(Xs + row * XS_STRIDE + col);
            asm volatile("global_load_async_to_lds_b128 %0, %1, off"
                         :: "v"(lds_off), "v"(gsrc) : "memory");
        }
    }
    asm volatile("s_wait_asynccnt 0" ::: "memory");
    __syncthreads();

    const int wv   = t >> 5;        // wave id 0..7 -> N tile
    const int lane = t & 31;
    const int half = lane >> 4;     // ISA: lanes 16-31 hold K+2,K+3 (A) / rows K+2,K+3 (B)
    const int mn   = lane & 15;     // M row (A) / N col (B,C,D)
    const int ncol = wv * 16 + mn;

    v8f c = {0.f, 0.f, 0.f, 0.f, 0.f, 0.f, 0.f, 0.f};

    #pragma unroll
    for (int k0 = 0; k0 < D_FEAT; k0 += 4) {
        const int k = k0 + 2 * half;
        // A: 16x4 tile of X from LDS (8B load, padded stride -> conflict-free)
        v2f a;
        a.x = Xs[mn * XS_STRIDE + k];
        a.y = Xs[mn * XS_STRIDE + k + 1];
        // B: 4x16 tile from transposed W -> one contiguous global_load_b64
        v2f b = *(const v2f*)(Wt + (size_t)ncol * D_FEAT + k);
        // (neg_a, A, neg_b, B, c_mod, C, reuse_a, reuse_b)
        c = __builtin_amdgcn_wmma_f32_16x16x4_f32(false, a, false, b,
                                                  (short)0, c, false, false);
    }

    // D layout: VGPR r -> row (r + 8*half), col = lane&15
    #pragma unroll
    for (int r = 0; r < 8; ++r) {
        int row = mBase + r + 8 * half;
        if (row < n) H[(size_t)row * D_FEAT + ncol] = c[r];
    }
}

// ---------------------------------------------------------------------------
// out = b + dinv[i]^2 * H[i]   (self-loop contribution; also zero-bases d_out)
// One thread per float4 of the output.
// ---------------------------------------------------------------------------
__global__ void out_init_kernel(const float* __restrict__ H,
                                const float* __restrict__ dinv,
                                const float* __restrict__ bias,
                                float* __restrict__ out, int n) {
    int t = blockIdx.x * blockDim.x + threadIdx.x;   // over n*32 float4 slots
    if (t >= n * 32) return;
    int node = t >> 5;
    int j4   = t & 31;
    float di = dinv[node];
    float s  = di * di;
    float4 h = ((const float4*)H)[t];
    float4 b = ((const float4*)bias)[j4];
    float4 o;
    o.x = fmaf(h.x, s, b.x);
    o.y = fmaf(h.y, s, b.y);
    o.z = fmaf(h.z, s, b.z);
    o.w = fmaf(h.w, s, b.w);
    ((float4*)out)[t] = o;
}

// ---------------------------------------------------------------------------
// One wave32 per edge: lane l handles out[dst][4l..4l+3] += w * H[src][4l..4l+3]
// Coalesced 512B float4 gather per edge (L2-resident H); 4 f32 atomics/lane.
// ---------------------------------------------------------------------------
__launch_bounds__(256)
__global__ void edge_scatter_kernel(const int* __restrict__ src,
                                    const int* __restrict__ dst,
                                    const float* __restrict__ H,
                                    const float* __restrict__ dinv,
                                    float* __restrict__ out, int e) {
    int gid = blockIdx.x * blockDim.x + threadIdx.x;
    int edge = gid >> 5;
    if (edge >= e) return;
    int lane = gid & 31;
    int s = src[edge];
    int d = dst[edge];
    float w = dinv[s] * dinv[d];
    float4 v = *(const float4*)(H + (size_t)s * D_FEAT + lane * 4);
    float* o = out + (size_t)d * D_FEAT + lane * 4;
    unsafeAtomicAdd(o + 0, v.x * w);
    unsafeAtomicAdd(o + 1, v.y * w);
    unsafeAtomicAdd(o + 2, v.z * w);
    unsafeAtomicAdd(o + 3, v.w * w);
}

extern "C" void kernel_launch(void* const* d_in, const int* in_sizes, int n_in,
                              void* d_out, int out_size, void* d_ws, size_t ws_size,
                              hipStream_t stream) {
    const float* X    = (const float*)d_in[0];   // [N,128] f32
    const int*   ei   = (const int*)d_in[1];     // [2,E]   int
    const float* W    = (const float*)d_in[2];   // [128,128] f32
    const float* bias = (const float*)d_in[3];   // [128] f32

    const int N = in_sizes[0] / D_FEAT;
    const int E = in_sizes[1] / 2;
    const int* src = ei;       // edge_index[0]
    const int* dst = ei + E;   // edge_index[1]

    // Workspace layout: H [N*128] f32, deg/dinv [N] f32, Wt [128*128] f32.
    float* H    = (float*)d_ws;
    float* dinv = H + (size_t)N * D_FEAT;
    float* Wt   = dinv + N;
    float* out  = (float*)d_out;

    deg_init_kernel <<<(N + 255) / 256, 256, 0, stream>>>(dinv, N);
    deg_count_kernel<<<(E + 255) / 256, 256, 0, stream>>>(dst, dinv, E);
    deg_rsqrt_kernel<<<(N + 255) / 256, 256, 0, stream>>>(dinv, N);

    w_transpose_kernel<<<(D_FEAT * D_FEAT) / 256, 256, 0, stream>>>(W, Wt);
    gcn_gemm_wmma_kernel<<<(N + 15) / 16, 256, 0, stream>>>(X, Wt, H, N);

    out_init_kernel<<<(N * 32 + 255) / 256, 256, 0, stream>>>(H, dinv, bias, out, N);
    edge_scatter_kernel<<<(E * 32 + 255) / 256, 256, 0, stream>>>(src, dst, H, dinv, out, E);
}